// SubConv_56126632624167
// MI455X (gfx1250) — compile-verified
//
#include <hip/hip_runtime.h>
#include <hip/hip_bf16.h>
#include <math.h>

// -----------------------------------------------------------------------------
// Batched 16x16 SPD matrix logarithm on gfx1250 (MI455X).
//   X = logm(S),  S = P[..., :16, :16],  P is (B,H,32,32) fp32 SPD.
// Method: inverse scaling-and-squaring, all-FP32-WMMA:
//   s = trace(S); B = S/s  (spectrum in (0,1])
//   4 levels of coupled Newton-Schulz sqrt  -> B^(1/16) = I + E
//   log(I+E) via degree-8 Horner series     -> L
//   X = 16*L + log(s)*I
// One wave32 per matrix; each 16x16x16 matmul = 4x v_wmma_f32_16x16x4_f32.
// -----------------------------------------------------------------------------

typedef __attribute__((ext_vector_type(2))) float v2f;
typedef __attribute__((ext_vector_type(8))) float v8f;

#define WAVES_PER_BLOCK 8
#define NS_LEVELS 4      // number of matrix square roots (2^4 = 16)
#define NS_ITERS  10     // Newton-Schulz iterations per level
#define LOG_TERMS 8      // degree of log(1+x) Taylor polynomial

// 16x16x16 fp32 matmul, A and B row-major 16x16 tiles in LDS.
// Returns C in the WMMA 16x16 f32 accumulator layout (8 VGPRs/lane).
__device__ __forceinline__ v8f mm16(const float* A, const float* B, int lane) {
  const int r  = lane & 15;        // A row / B column for this lane
  const int kh = (lane >> 4) << 1; // lanes 0-15 -> K pair {0,1}; 16-31 -> {2,3}
  v8f acc = {0.f, 0.f, 0.f, 0.f, 0.f, 0.f, 0.f, 0.f};
#pragma unroll
  for (int kc = 0; kc < 4; ++kc) {
    const int k0 = kc * 4 + kh;
    v2f a, b;
    // A fragment (16x4, MxK): lane holds row r, K = k0, k0+1
    a.x = A[r * 16 + k0];
    a.y = A[r * 16 + k0 + 1];
    // B fragment (4x16, KxN): lane holds column r, rows K = k0, k0+1
    b.x = B[(k0    ) * 16 + r];
    b.y = B[(k0 + 1) * 16 + r];
    acc = __builtin_amdgcn_wmma_f32_16x16x4_f32(
        /*neg_a=*/false, a, /*neg_b=*/false, b,
        /*c_mod=*/(short)0, acc, /*reuse_a=*/false, /*reuse_b=*/false);
  }
  return acc;
}

// Store accumulator tile to row-major LDS tile.
__device__ __forceinline__ void store_tile(float* D, v8f c, int lane) {
  const int n  = lane & 15;
  const int mh = (lane >> 4) * 8;
#pragma unroll
  for (int rr = 0; rr < 8; ++rr) D[(mh + rr) * 16 + n] = c[rr];
}

// Store D = scale*C + diagAdd*I  (fused diagonal shift, row-major LDS tile).
__device__ __forceinline__ void store_tile_diag(float* D, v8f c, float scale,
                                                float diagAdd, int lane) {
  const int n  = lane & 15;
  const int mh = (lane >> 4) * 8;
#pragma unroll
  for (int rr = 0; rr < 8; ++rr) {
    float v = scale * c[rr];
    if (mh + rr == n) v += diagAdd;
    D[(mh + rr) * 16 + n] = v;
  }
}

__global__ __launch_bounds__(WAVES_PER_BLOCK * 32)
void spd_logm16_kernel(const float* __restrict__ P, float* __restrict__ X,
                       int nMat) {
  // 4 wave-private 1KB tiles per wave: Y (holds S/B), Z, T, L
  __shared__ __align__(16) float lds[WAVES_PER_BLOCK * 4 * 256];
  const int lane = threadIdx.x & 31;
  const int wave = threadIdx.x >> 5;
  const int mat  = blockIdx.x * WAVES_PER_BLOCK + wave;
  if (mat >= nMat) return;  // wave-uniform: EXEC stays all-ones for WMMA

  float* Y = &lds[wave * 1024];
  float* Z = Y + 256;
  float* T = Y + 512;
  float* L = Y + 768;

  // ---- Load top-left 16x16 of this 32x32 matrix (2x b128 per lane) ----
  const float* src = P + (size_t)mat * 1024;
  const int row = lane >> 1;
  const int c0  = (lane & 1) * 8;
  float4 g0 = *(const float4*)(src + row * 32 + c0);
  float4 g1 = *(const float4*)(src + row * 32 + c0 + 4);
  *(float4*)(Y + row * 16 + c0)     = g0;
  *(float4*)(Y + row * 16 + c0 + 4) = g1;

  // ---- trace -> spectral scaling (trace >= lambda_max since SPD) ----
  float tr = 0.f;
#pragma unroll
  for (int i = 0; i < 16; ++i) tr += Y[i * 17];
  tr = fmaxf(tr, 1e-12f);
  const float inv_s = 1.0f / tr;
  const float log_s = logf(tr);

  // B = S/s in Y;  Z = I
#pragma unroll
  for (int j = 0; j < 8; ++j) Y[row * 16 + c0 + j] *= inv_s;
  {
    const int n = lane & 15, mh = (lane >> 4) * 8;
#pragma unroll
    for (int rr = 0; rr < 8; ++rr)
      Z[(mh + rr) * 16 + n] = (mh + rr == n) ? 1.f : 0.f;
  }

  // ---- NS_LEVELS coupled Newton-Schulz square roots: Y <- Y^(1/2) ----
#pragma unroll 1
  for (int lev = 0; lev < NS_LEVELS; ++lev) {
#pragma unroll 1
    for (int it = 0; it < NS_ITERS; ++it) {
      v8f m = mm16(Z, Y, lane);                  // M = Z*Y
      store_tile_diag(T, m, -0.5f, 1.5f, lane);  // T = 1.5I - 0.5M
      v8f y = mm16(Y, T, lane);                  // Y*T
      v8f z = mm16(T, Z, lane);                  // T*Z
      store_tile(Y, y, lane);
      store_tile(Z, z, lane);
    }
    const int n = lane & 15, mh = (lane >> 4) * 8;
#pragma unroll
    for (int rr = 0; rr < 8; ++rr)               // reset Z = I for next level
      Z[(mh + rr) * 16 + n] = (mh + rr == n) ? 1.f : 0.f;
  }

  // ---- E = Y - I (into T);  L = c_K * I ----
  {
    const float cK = ((LOG_TERMS & 1) ? 1.0f : -1.0f) / (float)LOG_TERMS;
    const int n = lane & 15, mh = (lane >> 4) * 8;
#pragma unroll
    for (int rr = 0; rr < 8; ++rr) {
      const int mI = mh + rr;
      const float e = Y[mI * 16 + n] - ((mI == n) ? 1.f : 0.f);
      T[mI * 16 + n] = e;
      L[mI * 16 + n] = (mI == n) ? cK : 0.f;
    }
  }

  // ---- Horner: L <- c_j*I + E*L,  j = K-1 .. 1 ----
#pragma unroll 1
  for (int j = LOG_TERMS - 1; j >= 1; --j) {
    const float cj = ((j & 1) ? 1.0f : -1.0f) / (float)j;
    v8f p = mm16(T, L, lane);
    store_tile_diag(L, p, 1.0f, cj, lane);
  }

  // ---- X = 2^NS_LEVELS * (E*L) + log(s)*I  -> reuse Y tile ----
  {
    v8f p = mm16(T, L, lane);
    store_tile_diag(Y, p, (float)(1 << NS_LEVELS), log_s, lane);
  }

  // ---- Coalesced write-out (16x16 fp32 per matrix) ----
  float* dst = X + (size_t)mat * 256;
  float4 o0 = *(const float4*)(Y + row * 16 + c0);
  float4 o1 = *(const float4*)(Y + row * 16 + c0 + 4);
  *(float4*)(dst + row * 16 + c0)     = o0;
  *(float4*)(dst + row * 16 + c0 + 4) = o1;
}

extern "C" void kernel_launch(void* const* d_in, const int* in_sizes, int n_in,
                              void* d_out, int out_size, void* d_ws,
                              size_t ws_size, hipStream_t stream) {
  (void)n_in; (void)out_size; (void)d_ws; (void)ws_size;
  const float* P = (const float*)d_in[0];
  // d_in[1] is k (==16); kernel is specialized for the 16x16 WMMA tile.
  const int nMat = in_sizes[0] / 1024;  // BATCH*H matrices of 32x32 floats
  dim3 block(WAVES_PER_BLOCK * 32);
  dim3 grid((nMat + WAVES_PER_BLOCK - 1) / WAVES_PER_BLOCK);
  spd_logm16_kernel<<<grid, block, 0, stream>>>(P, (float*)d_out, nMat);
}